// Criterion_12180527252198
// MI455X (gfx1250) — compile-verified
//
#include <hip/hip_runtime.h>
#include <hip/hip_bf16.h>
#include <cfloat>

#define DEV_INLINE __device__ __forceinline__

// Problem constants (match reference)
constexpr int   BB     = 32;
constexpr int   MM     = 8192;
constexpr int   GG     = 32;
constexpr int   CC     = 80;     // NUM_CLASSES
constexpr float IGT    = 0.7f;
constexpr float IOU_T  = 0.15f;
constexpr float ALPHA  = 0.25f;

// grids
constexpr int ASSIGN_BLOCKS = BB * GG;          // 1024
constexpr int MAXIOU_BLOCKS = BB * (MM / 256);  // 1024
constexpr int FOCAL_BLOCKS  = 4096;
constexpr int N4            = BB * MM * CC / 4; // 5,242,880 (= 5 * 4096*256)

typedef __attribute__((ext_vector_type(2))) float v2f;
typedef __attribute__((ext_vector_type(8))) float v8f;

DEV_INLINE float iou_xyxy(float4 a, float4 b) {
  float areaA = (a.z - a.x) * (a.w - a.y);
  float areaB = (b.z - b.x) * (b.w - b.y);
  float iw = fmaxf(fminf(a.z, b.z) - fmaxf(a.x, b.x), 0.f);
  float ih = fmaxf(fminf(a.w, b.w) - fmaxf(a.y, b.y), 0.f);
  float inter = iw * ih;
  float uni = areaA + areaB - inter;
  return inter / fmaxf(uni, 1e-9f);
}

DEV_INLINE float giou_xyxy(float4 a, float4 b) {
  float areaA = (a.z - a.x) * (a.w - a.y);
  float areaB = (b.z - b.x) * (b.w - b.y);
  float iw = fmaxf(fminf(a.z, b.z) - fmaxf(a.x, b.x), 0.f);
  float ih = fmaxf(fminf(a.w, b.w) - fmaxf(a.y, b.y), 0.f);
  float inter = iw * ih;
  float uni = areaA + areaB - inter;
  float iou = inter / fmaxf(uni, 1e-9f);
  float ew = fmaxf(fmaxf(a.z, b.z) - fminf(a.x, b.x), 0.f);
  float eh = fmaxf(fmaxf(a.w, b.w) - fminf(a.y, b.y), 0.f);
  float areaE = ew * eh;
  return iou - (areaE - uni) / fmaxf(areaE, 1e-9f);
}

DEV_INLINE void top4_insert(float v, int id,
                            float& c0, float& c1, float& c2, float& c3,
                            int& i0, int& i1, int& i2, int& i3) {
  if (v < c3) {
    c3 = v; i3 = id;
    if (c3 < c2) { float tc = c2; c2 = c3; c3 = tc; int ti = i2; i2 = i3; i3 = ti; }
    if (c2 < c1) { float tc = c1; c1 = c2; c2 = tc; int ti = i1; i1 = i2; i2 = ti; }
    if (c1 < c0) { float tc = c0; c0 = c1; c1 = tc; int ti = i0; i0 = i1; i1 = ti; }
  }
}

// Merge two sorted-ascending 4-lists (LDS resident) -> 4 smallest into dst slots.
DEV_INLINE void merge_top4(float* c, int* idx, int dst, int src) {
  int ai = 0, bi = 0;
  float rc[4]; int ri[4];
#pragma unroll
  for (int k = 0; k < 4; ++k) {
    float av = c[dst * 4 + ai], bv = c[src * 4 + bi];
    bool ta = av <= bv;
    rc[k] = ta ? av : bv;
    ri[k] = ta ? idx[dst * 4 + ai] : idx[src * 4 + bi];
    if (ta) ++ai; else ++bi;
  }
#pragma unroll
  for (int k = 0; k < 4; ++k) { c[dst * 4 + k] = rc[k]; idx[dst * 4 + k] = ri[k]; }
}

// ---------------------------------------------------------------------------
// Kernel A: per (b,g) -> top-4 by pred cost and by anchor cost; emit matched
// src indices, matched labels (with pos-ignore), and per-block GIoU partial.
// ---------------------------------------------------------------------------
__global__ __launch_bounds__(256) void assign_kernel(
    const float4* __restrict__ pred_box, const float4* __restrict__ anchors,
    const float4* __restrict__ tgt_boxes, const int* __restrict__ tgt_labels,
    int* __restrict__ srcA, int* __restrict__ tgtA, float* __restrict__ regP) {
  __shared__ float sPc[1024]; __shared__ int sPi[1024];
  __shared__ float sAc[1024]; __shared__ int sAi[1024];
  __shared__ float sContrib[8];
  const int tid = threadIdx.x;
  const int b = blockIdx.x >> 5, g = blockIdx.x & 31;
  const float4 t = tgt_boxes[b * GG + g];

  float pc0 = FLT_MAX, pc1 = FLT_MAX, pc2 = FLT_MAX, pc3 = FLT_MAX;
  int   pi0 = 0, pi1 = 0, pi2 = 0, pi3 = 0;
  float ac0 = FLT_MAX, ac1 = FLT_MAX, ac2 = FLT_MAX, ac3 = FLT_MAX;
  int   ai0 = 0, ai1 = 0, ai2 = 0, ai3 = 0;

  for (int it = 0; it < MM / 256; ++it) {
    const int m = tid + (it << 8);
    float4 pb = pred_box[b * MM + m];
    float cp = fabsf(pb.x - t.x) + fabsf(pb.y - t.y) + fabsf(pb.z - t.z) + fabsf(pb.w - t.w);
    top4_insert(cp, m, pc0, pc1, pc2, pc3, pi0, pi1, pi2, pi3);

    float4 a = anchors[m];
    float ax0 = a.x - 0.5f * a.z, ay0 = a.y - 0.5f * a.w;
    float ax1 = a.x + 0.5f * a.z, ay1 = a.y + 0.5f * a.w;
    float ca = fabsf(ax0 - t.x) + fabsf(ay0 - t.y) + fabsf(ax1 - t.z) + fabsf(ay1 - t.w);
    top4_insert(ca, m, ac0, ac1, ac2, ac3, ai0, ai1, ai2, ai3);
  }

  sPc[tid * 4 + 0] = pc0; sPc[tid * 4 + 1] = pc1; sPc[tid * 4 + 2] = pc2; sPc[tid * 4 + 3] = pc3;
  sPi[tid * 4 + 0] = pi0; sPi[tid * 4 + 1] = pi1; sPi[tid * 4 + 2] = pi2; sPi[tid * 4 + 3] = pi3;
  sAc[tid * 4 + 0] = ac0; sAc[tid * 4 + 1] = ac1; sAc[tid * 4 + 2] = ac2; sAc[tid * 4 + 3] = ac3;
  sAi[tid * 4 + 0] = ai0; sAi[tid * 4 + 1] = ai1; sAi[tid * 4 + 2] = ai2; sAi[tid * 4 + 3] = ai3;
  __syncthreads();

  for (int stride = 128; stride > 0; stride >>= 1) {
    if (tid < stride) {
      merge_top4(sPc, sPi, tid, tid + stride);
      merge_top4(sAc, sAi, tid, tid + stride);
    }
    __syncthreads();
  }

  if (tid < 8) {
    const int src = (tid < 4) ? sPi[tid] : sAi[tid - 4];
    float4 a = anchors[src];
    float4 axy = make_float4(a.x - 0.5f * a.z, a.y - 0.5f * a.w,
                             a.x + 0.5f * a.z, a.y + 0.5f * a.w);
    float aiou = iou_xyxy(axy, t);
    bool posIgn = aiou < IOU_T;
    int lbl = posIgn ? -1 : tgt_labels[b * GG + g];
    int j = (g << 3) + tid;                 // position within [0, G*2K)
    srcA[b * 256 + j] = src;
    tgtA[b * 256 + j] = lbl;
    float4 pb = pred_box[b * MM + src];
    float gv = giou_xyxy(pb, t);
    sContrib[tid] = posIgn ? 0.f : (1.f - gv);
  }
  __syncthreads();
  if (tid == 0) {
    float s = 0.f;
#pragma unroll
    for (int k = 0; k < 8; ++k) s += sContrib[k];
    regP[blockIdx.x] = s;
  }
}

// ---------------------------------------------------------------------------
// Kernel B: per (b,m) max IoU vs all G targets -> gt_cls init (-1 ignore / C bg)
// ---------------------------------------------------------------------------
__global__ __launch_bounds__(256) void maxiou_kernel(
    const float4* __restrict__ pred_box, const float4* __restrict__ tgt_boxes,
    int* __restrict__ gt_cls) {
  __shared__ float4 tb[GG];
  const int tid = threadIdx.x;
  const int b = blockIdx.x >> 5;
  const int m = ((blockIdx.x & 31) << 8) + tid;
  if (tid < GG) tb[tid] = tgt_boxes[b * GG + tid];
  __syncthreads();
  float4 pb = pred_box[b * MM + m];
  float mx = 0.f;
#pragma unroll 4
  for (int g = 0; g < GG; ++g) mx = fmaxf(mx, iou_xyxy(pb, tb[g]));
  gt_cls[b * MM + m] = (mx > IGT) ? -1 : CC;
}

// ---------------------------------------------------------------------------
// Kernel C: ordered scatter (serialized per batch -> deterministic last-write)
// ---------------------------------------------------------------------------
__global__ void scatter_kernel(const int* __restrict__ srcA,
                               const int* __restrict__ tgtA,
                               int* __restrict__ gt_cls) {
  const int b = threadIdx.x;
  if (b < BB) {
    for (int j = 0; j < 256; ++j)
      gt_cls[b * MM + srcA[b * 256 + j]] = tgtA[b * 256 + j];
  }
}

// ---------------------------------------------------------------------------
// Kernel D: streaming focal loss over 84MB of logits (float4) + fg count.
// Wave-level reduction via V_WMMA_F32_16X16X4_F32 (B = ones => row-sums).
// ---------------------------------------------------------------------------
DEV_INLINE float focal1(float x, bool pos) {
  float t  = pos ? 1.f : 0.f;
  float ce = fmaxf(x, 0.f) - x * t + __logf(1.f + __expf(-fabsf(x)));
  float p  = 1.f / (1.f + __expf(-x));
  float pt = pos ? p : (1.f - p);
  float at = pos ? ALPHA : (1.f - ALPHA);
  float om = 1.f - pt;
  return at * ce * om * om;
}

__global__ __launch_bounds__(256) void focal_kernel(
    const float4* __restrict__ pcls4, const int* __restrict__ gt_cls,
    const unsigned char* __restrict__ mask,
    float* __restrict__ clsP, int* __restrict__ fgP) {
  const int tid = threadIdx.x;
  float sum = 0.f;
  int fg = 0;
  for (int i = blockIdx.x * 256 + tid; i < N4; i += gridDim.x * 256) {
    float4 x4 = pcls4[i];
    int row = i / 20;           // 20 float4 groups per row of 80 classes
    int q = i - row * 20;
    int gt = gt_cls[row];
    if (q == 0) fg += (gt >= 0 && gt < CC) ? 1 : 0;   // one count per row
    if (gt >= 0 && mask[row]) {
      int c0 = q << 2;
      sum += focal1(x4.x, c0 + 0 == gt);
      sum += focal1(x4.y, c0 + 1 == gt);
      sum += focal1(x4.z, c0 + 2 == gt);
      sum += focal1(x4.w, c0 + 3 == gt);
    }
  }

  // --- wave32 reduction of `sum` via WMMA: A(VGPR0)=partials, A(VGPR1)=0,
  // B=ones => D[m,n] = s_m + s_{m+16}; sum the 8 D VGPRs, then xor-16 shuffle.
  v2f a;  a[0] = sum; a[1] = 0.f;
  v2f bo; bo[0] = 1.f; bo[1] = 1.f;
  v8f cz = {};
  v8f d = __builtin_amdgcn_wmma_f32_16x16x4_f32(false, a, false, bo,
                                                (short)0, cz, false, false);
  float w = d[0] + d[1] + d[2] + d[3] + d[4] + d[5] + d[6] + d[7];
  w += __shfl_xor(w, 16);                 // halves -> full 32-lane sum

#pragma unroll
  for (int off = 16; off > 0; off >>= 1) fg += __shfl_xor(fg, off);

  __shared__ float wsum[8];
  __shared__ int   wfg[8];
  const int lane = tid & 31, wid = tid >> 5;
  if (lane == 0) { wsum[wid] = w; wfg[wid] = fg; }
  __syncthreads();
  if (tid == 0) {
    float s = 0.f; int f = 0;
#pragma unroll
    for (int k = 0; k < 8; ++k) { s += wsum[k]; f += wfg[k]; }
    clsP[blockIdx.x] = s;
    fgP[blockIdx.x] = f;
  }
}

// ---------------------------------------------------------------------------
// Kernel F: deterministic final reduction + normalization -> 3 outputs
// ---------------------------------------------------------------------------
__global__ __launch_bounds__(256) void finalize_kernel(
    const float* __restrict__ clsP, const int* __restrict__ fgP,
    const float* __restrict__ regP, float* __restrict__ out) {
  __shared__ float sF[256]; __shared__ int sI[256]; __shared__ float sR[256];
  const int tid = threadIdx.x;
  float cs = 0.f; int f = 0; float rs = 0.f;
  for (int i = tid; i < FOCAL_BLOCKS; i += 256) { cs += clsP[i]; f += fgP[i]; }
  for (int i = tid; i < ASSIGN_BLOCKS; i += 256) rs += regP[i];
  sF[tid] = cs; sI[tid] = f; sR[tid] = rs;
  __syncthreads();
  for (int s = 128; s > 0; s >>= 1) {
    if (tid < s) { sF[tid] += sF[tid + s]; sI[tid] += sI[tid + s]; sR[tid] += sR[tid + s]; }
    __syncthreads();
  }
  if (tid == 0) {
    float nf = fmaxf((float)sI[0], 1.f);
    float lc = sF[0] / nf, lr = sR[0] / nf;
    out[0] = lc; out[1] = lr; out[2] = lc + lr;
  }
}

// ---------------------------------------------------------------------------
extern "C" void kernel_launch(void* const* d_in, const int* in_sizes, int n_in,
                              void* d_out, int out_size, void* d_ws, size_t ws_size,
                              hipStream_t stream) {
  const float4* pred_cls4 = (const float4*)d_in[0];         // (B,M,C) f32
  const float4* pred_box  = (const float4*)d_in[1];         // (B,M,4) f32
  const float4* anchors   = (const float4*)d_in[2];         // (M,4)   f32 cxcywh
  const unsigned char* mask = (const unsigned char*)d_in[3];// (B*M,)  bool
  const float4* tgt_boxes = (const float4*)d_in[4];         // (B,G,4) f32
  const int* tgt_labels   = (const int*)d_in[5];            // (B,G)   int
  float* out = (float*)d_out;

  char* ws = (char*)d_ws;
  float* clsP = (float*)(ws + 0);        // FOCAL_BLOCKS  floats (16 KB)
  int*   fgP  = (int*)(ws + 16384);      // FOCAL_BLOCKS  ints   (16 KB)
  float* regP = (float*)(ws + 32768);    // ASSIGN_BLOCKS floats ( 4 KB)
  int*   srcA = (int*)(ws + 36864);      // B*256 ints (32 KB)
  int*   tgtA = (int*)(ws + 69632);      // B*256 ints (32 KB)
  int*   gtC  = (int*)(ws + 102400);     // B*M ints   ( 1 MB)

  assign_kernel<<<ASSIGN_BLOCKS, 256, 0, stream>>>(pred_box, anchors, tgt_boxes,
                                                   tgt_labels, srcA, tgtA, regP);
  maxiou_kernel<<<MAXIOU_BLOCKS, 256, 0, stream>>>(pred_box, tgt_boxes, gtC);
  scatter_kernel<<<1, 32, 0, stream>>>(srcA, tgtA, gtC);
  focal_kernel<<<FOCAL_BLOCKS, 256, 0, stream>>>(pred_cls4, gtC, mask, clsP, fgP);
  finalize_kernel<<<1, 256, 0, stream>>>(clsP, fgP, regP, out);
}